// ThLista_78975858639048
// MI455X (gfx1250) — compile-verified
//
#include <hip/hip_runtime.h>
#include <math.h>

// Problem constants (match reference)
#define NN 1024   // grid size / rows of x
#define MM 256    // measurements
#define BB 512    // batch
#define LL 10     // layers
#define BETA 0.01f
#define VLEN (2 * NN - 1)   // 2047, Toeplitz generator length
#define APITCH 18           // LDS row pitch (even -> 8B-aligned b64, conflict-free)

#define ABUF (64 * APITCH)               // floats per A plane
#define BBUFS (32 * APITCH)              // floats per B plane
#define STAGE (2 * ABUF + 2 * BBUFS)     // floats per pipeline stage
#define ASYNC_PER_STAGE 24               // async b32 loads issued per thread per stage

typedef __attribute__((ext_vector_type(2))) float v2f;
typedef __attribute__((ext_vector_type(8))) float v8f;

// v1 = concat(conj(flip(v)), v[1:])  for every layer
__global__ void build_v1_kernel(const float* __restrict__ v_re, const float* __restrict__ v_im,
                                float* __restrict__ v1_re, float* __restrict__ v1_im) {
    int t = blockIdx.x * blockDim.x + threadIdx.x;
    if (t >= LL * VLEN) return;
    int l = t / VLEN;
    int j = t - l * VLEN;
    int src = (j <= NN - 1) ? (NN - 1 - j) : (j - (NN - 1));
    float re = v_re[l * NN + src];
    float im = v_im[l * NN + src];
    if (j <= NN - 1) im = -im;          // conjugate half
    v1_re[t] = re;
    v1_im[t] = im;
}

// gfx1250 async global->LDS staging (GVS mode: SGPR64 base + per-lane i32 byte offset).
// Tracked by ASYNCcnt; async loads complete in order.
__device__ __forceinline__ void async_ld_b32(unsigned lds_off, unsigned gbyte_off, const void* sbase) {
    asm volatile("global_load_async_to_lds_b32 %0, %1, %2"
                 :: "v"(lds_off), "v"(gbyte_off), "s"(sbase) : "memory");
}
template <int N>
__device__ __forceinline__ void wait_async_le() {
    asm volatile("s_wait_asynccnt %0" :: "i"(N) : "memory");
}

// One LISTA layer: out = soft_thresh(W1 @ x + W2 @ y, BETA)
// W1[r][k] = v1[r - k + N - 1] (Hermitian Toeplitz, gathered on the fly).
// Block: 128 threads = 4 wave32. Block tile 64x32; wave tile 16 rows x 32 cols.
// Double-buffered LDS: chunk k+1 staged async while chunk k feeds the WMMAs.
__launch_bounds__(128)
__global__ void lista_layer_kernel(const float* __restrict__ v1_re, const float* __restrict__ v1_im,
                                   const float* __restrict__ w2_re, const float* __restrict__ w2_im,
                                   const float* __restrict__ y_re,  const float* __restrict__ y_im,
                                   const float* __restrict__ x_re,  const float* __restrict__ x_im,
                                   float* __restrict__ o_re, float* __restrict__ o_im,
                                   int has_x, int interleave)
{
    __shared__ __align__(16) float smem[2 * STAGE];

    const int tid  = threadIdx.x;
    const int lane = tid & 31;
    const int w    = tid >> 5;            // wave id 0..3 (wave32)
    const int row0 = blockIdx.x * 64;     // output row block
    const int col0 = blockIdx.y * 32;     // output col block

    const int mrow  = w * 16;             // wave's 16-row tile inside block
    const int mi    = lane & 15;
    const int khalf = (lane >> 4) * 2;    // lanes 0-15 -> K 0,1 ; lanes 16-31 -> K 2,3

    v8f acc_re0 = {}, acc_im0 = {};       // col tile 0 (cols 0..15)
    v8f acc_re1 = {}, acc_im1 = {};       // col tile 1 (cols 16..31)

    // Stage one 16-wide k chunk into LDS stage buffer `sb` with async loads.
    // Chunks never straddle k==NN (NN % 16 == 0) -> source select is wave-uniform,
    // so the base pointer stays in SGPRs (GVS addressing).
    auto stage = [&](int kc, float* sb) {
        const bool w1p = (kc < NN);
        float* sAre = sb;
        float* sAim = sb + ABUF;
        float* sBre = sb + 2 * ABUF;
        float* sBim = sb + 2 * ABUF + BBUFS;
        // A tile: 64 rows x 16 k (complex)
        #pragma unroll
        for (int i = 0; i < 8; ++i) {
            int idx = tid + i * 128;          // 0..1023
            int r   = idx >> 4;
            int kk  = idx & 15;
            int k   = kc + kk;
            unsigned lre = (unsigned)(uintptr_t)&sAre[r * APITCH + kk];
            unsigned lim = (unsigned)(uintptr_t)&sAim[r * APITCH + kk];
            if (w1p) {                        // Toeplitz gather from v1
                unsigned go = (unsigned)((row0 + r - k + (NN - 1)) * 4);
                async_ld_b32(lre, go, v1_re);
                async_ld_b32(lim, go, v1_im);
            } else {                          // dense W2 block, coalesced in k
                unsigned go = (unsigned)(((row0 + r) * MM + (k - NN)) * 4);
                async_ld_b32(lre, go, w2_re);
                async_ld_b32(lim, go, w2_im);
            }
        }
        // B tile: 16 k x 32 cols (complex), stored transposed [c][k]
        #pragma unroll
        for (int i = 0; i < 4; ++i) {
            int idx = tid + i * 128;          // 0..511
            int c   = idx & 31;
            int kk  = idx >> 5;               // 0..15
            int k   = kc + kk;
            unsigned lre = (unsigned)(uintptr_t)&sBre[c * APITCH + kk];
            unsigned lim = (unsigned)(uintptr_t)&sBim[c * APITCH + kk];
            if (w1p) {                        // from x, coalesced in c
                unsigned go = (unsigned)((k * BB + col0 + c) * 4);
                async_ld_b32(lre, go, x_re);
                async_ld_b32(lim, go, x_im);
            } else {                          // from y
                unsigned go = (unsigned)(((k - NN) * BB + col0 + c) * 4);
                async_ld_b32(lre, go, y_re);
                async_ld_b32(lim, go, y_im);
            }
        }
    };

    const int k_begin = has_x ? 0 : NN;   // layer 0: x == 0, only W2@y part
    const int k_end   = NN + MM;

    // Prologue: stage first chunk.
    stage(k_begin, smem + ((k_begin >> 4) & 1) * STAGE);

    for (int k0 = k_begin; k0 < k_end; k0 += 16) {
        const int cur = (k0 >> 4) & 1;
        const int kn  = k0 + 16;
        if (kn < k_end) {
            // Prefetch next chunk into the other buffer, then wait only until the
            // CURRENT chunk's 24 loads are done (in-order completion): the next
            // chunk's latency overlaps this chunk's WMMAs.
            stage(kn, smem + (cur ^ 1) * STAGE);
            wait_async_le<ASYNC_PER_STAGE>();
        } else {
            wait_async_le<0>();
        }
        __syncthreads();      // everyone's chunk-k data landed in LDS

        const float* sb   = smem + cur * STAGE;
        const float* sAre = sb;
        const float* sAim = sb + ABUF;
        const float* sBre = sb + 2 * ABUF;
        const float* sBim = sb + 2 * ABUF + BBUFS;

        // 16 k-values = 4 fp32 WMMA k-steps; complex = 4 real WMMAs per C tile.
        #pragma unroll
        for (int kk = 0; kk < 16; kk += 4) {
            v2f ar  = *(const v2f*)&sAre[(mrow + mi) * APITCH + kk + khalf];
            v2f ai  = *(const v2f*)&sAim[(mrow + mi) * APITCH + kk + khalf];
            v2f br0 = *(const v2f*)&sBre[(mi)      * APITCH + kk + khalf];
            v2f bi0 = *(const v2f*)&sBim[(mi)      * APITCH + kk + khalf];
            v2f br1 = *(const v2f*)&sBre[(16 + mi) * APITCH + kk + khalf];
            v2f bi1 = *(const v2f*)&sBim[(16 + mi) * APITCH + kk + khalf];
            v2f nai = -ai;  // fp32 WMMA NEG field only supports CNeg -> negate A in VALU

            acc_re0 = __builtin_amdgcn_wmma_f32_16x16x4_f32(false, ar,  false, br0, (short)0, acc_re0, false, false);
            acc_re0 = __builtin_amdgcn_wmma_f32_16x16x4_f32(false, nai, false, bi0, (short)0, acc_re0, false, false);
            acc_im0 = __builtin_amdgcn_wmma_f32_16x16x4_f32(false, ar,  false, bi0, (short)0, acc_im0, false, false);
            acc_im0 = __builtin_amdgcn_wmma_f32_16x16x4_f32(false, ai,  false, br0, (short)0, acc_im0, false, false);

            acc_re1 = __builtin_amdgcn_wmma_f32_16x16x4_f32(false, ar,  false, br1, (short)0, acc_re1, false, false);
            acc_re1 = __builtin_amdgcn_wmma_f32_16x16x4_f32(false, nai, false, bi1, (short)0, acc_re1, false, false);
            acc_im1 = __builtin_amdgcn_wmma_f32_16x16x4_f32(false, ar,  false, bi1, (short)0, acc_im1, false, false);
            acc_im1 = __builtin_amdgcn_wmma_f32_16x16x4_f32(false, ai,  false, br1, (short)0, acc_im1, false, false);
        }
        __syncthreads();      // all waves done reading buf `cur` before it is re-staged
    }

    // ---- epilogue: complex soft-threshold + store ----
    // C/D layout: VGPR i -> row i (lanes 0-15) / row i+8 (lanes 16-31), col = lane%16
    const int rbase = row0 + mrow + ((lane >> 4) << 3);
    const int c0    = col0 + mi;
    const int c1    = col0 + 16 + mi;
    #pragma unroll
    for (int i = 0; i < 8; ++i) {
        int row = rbase + i;
        {
            float re  = acc_re0[i];
            float im  = acc_im0[i];
            float mag = sqrtf(re * re + im * im);
            float s   = fmaxf(mag - BETA, 0.0f) / fmaxf(mag, 1e-30f);
            re *= s; im *= s;
            size_t o = (size_t)row * BB + c0;
            if (interleave) { o_re[2 * o] = re; o_re[2 * o + 1] = im; }
            else            { o_re[o] = re; o_im[o] = im; }
        }
        {
            float re  = acc_re1[i];
            float im  = acc_im1[i];
            float mag = sqrtf(re * re + im * im);
            float s   = fmaxf(mag - BETA, 0.0f) / fmaxf(mag, 1e-30f);
            re *= s; im *= s;
            size_t o = (size_t)row * BB + c1;
            if (interleave) { o_re[2 * o] = re; o_re[2 * o + 1] = im; }
            else            { o_re[o] = re; o_im[o] = im; }
        }
    }
}

extern "C" void kernel_launch(void* const* d_in, const int* in_sizes, int n_in,
                              void* d_out, int out_size, void* d_ws, size_t ws_size,
                              hipStream_t stream) {
    (void)in_sizes; (void)n_in; (void)out_size; (void)ws_size;

    const float* v_re  = (const float*)d_in[0];
    const float* v_im  = (const float*)d_in[1];
    const float* W2_re = (const float*)d_in[2];
    const float* W2_im = (const float*)d_in[3];
    const float* y_re  = (const float*)d_in[4];
    const float* y_im  = (const float*)d_in[5];

    // Workspace layout (floats): v1 planes, then ping buffer A (planar x).
    float* ws    = (float*)d_ws;
    float* v1_re = ws;
    float* v1_im = v1_re + (size_t)LL * VLEN;
    float* xA_re = v1_im + (size_t)LL * VLEN;
    float* xA_im = xA_re + (size_t)NN * BB;
    // Buffer B reuses d_out as planar scratch (4 MB = exactly re+im planes).
    // Safe: the final layer reads only buffer A and overwrites d_out interleaved.
    float* outf  = (float*)d_out;
    float* xB_re = outf;
    float* xB_im = outf + (size_t)NN * BB;

    {
        int total = LL * VLEN;
        build_v1_kernel<<<(total + 255) / 256, 256, 0, stream>>>(v_re, v_im, v1_re, v1_im);
    }

    dim3 grid(NN / 64, BB / 32);   // 16 x 16 = 256 workgroups
    dim3 block(128);               // 4 wave32 waves

    // Layer 0: x = 0 -> only W2@y. Writes buffer A.
    lista_layer_kernel<<<grid, block, 0, stream>>>(v1_re, v1_im, W2_re, W2_im, y_re, y_im,
                                                   nullptr, nullptr, xA_re, xA_im, 0, 0);

    // Layers 1..9 ping-pong A <-> B; layer 9 (odd, reads A) writes d_out interleaved.
    for (int l = 1; l < LL; ++l) {
        const float* v1r = v1_re + (size_t)l * VLEN;
        const float* v1i = v1_im + (size_t)l * VLEN;
        const float* w2r = W2_re + (size_t)l * NN * MM;
        const float* w2i = W2_im + (size_t)l * NN * MM;
        const float* xr_in = (l & 1) ? xA_re : xB_re;
        const float* xi_in = (l & 1) ? xA_im : xB_im;
        if (l < LL - 1) {
            float* xr_out = (l & 1) ? xB_re : xA_re;
            float* xi_out = (l & 1) ? xB_im : xA_im;
            lista_layer_kernel<<<grid, block, 0, stream>>>(v1r, v1i, w2r, w2i, y_re, y_im,
                                                           xr_in, xi_in, xr_out, xi_out, 1, 0);
        } else {
            lista_layer_kernel<<<grid, block, 0, stream>>>(v1r, v1i, w2r, w2i, y_re, y_im,
                                                           xr_in, xi_in, outf, nullptr, 1, 1);
        }
    }
}